// Maximum_Path_Generater_40149354283339
// MI455X (gfx1250) — compile-verified
//
#include <hip/hip_runtime.h>
#include <stdint.h>

// Problem shape (from reference setup_inputs)
constexpr int B = 32, T = 384, M = 1536;
constexpr int K = T / 32;       // rows per lane (strided row ownership), 12
constexpr int JC = M / 32;      // decision-bit column chunks, 48
constexpr int TILE_C = 4;       // columns staged per TDM tile

#define NEGV (-1.0e7f)

typedef unsigned int __attribute__((ext_vector_type(4))) v4u;
typedef int   __attribute__((ext_vector_type(4))) v4i;
typedef int   __attribute__((ext_vector_type(8))) v8i;
typedef float __attribute__((ext_vector_type(2))) v2f;
typedef float __attribute__((ext_vector_type(8))) v8f;

// ---------------------------------------------------------------------------
// TDM: stage a [T rows x TILE_C cols] f32 tile (row stride = M elements) from
// global into contiguous LDS [T][TILE_C]. Group-1 (shape/strides) is loop
// invariant and passed in prebuilt; only the 3 address words vary per issue.
// ---------------------------------------------------------------------------
__device__ __forceinline__ v8i tdm_make_g1() {
  v8i g1;
  g1[0] = (2 << 16);                                          // data_size=4B, wg_mask=0
  g1[1] = (int)((M & 0xFFFF) << 16);                          // tensor_dim0[15:0]
  g1[2] = (int)(((T & 0xFFFF) << 16) | ((M >> 16) & 0xFFFF)); // dim1 lo | dim0 hi
  g1[3] = (int)((TILE_C << 16) | ((T >> 16) & 0xFFFF));       // tile_dim0 | dim1 hi
  g1[4] = T;                                                  // tile_dim1=T, tile_dim2=0
  g1[5] = M;                                                  // tensor_dim0_stride
  g1[6] = 0;
  g1[7] = 0;
  return g1;
}

__device__ __forceinline__ void tdm_issue(const float* gsrc, uint32_t lds_off, v8i g1) {
  uint64_t ga = (uint64_t)(uintptr_t)gsrc;
  v4u g0;
  g0[0] = 1u;                                                // count=1, user mode
  g0[1] = lds_off;                                           // lds_addr (bytes)
  g0[2] = (uint32_t)ga;                                      // global_addr[31:0]
  g0[3] = (uint32_t)((ga >> 32) & 0x1FFFFFFu) | 0x80000000u; // addr[56:32] | type=2
  v4i z4 = {0, 0, 0, 0};
#if __has_include(<hip/amd_detail/amd_gfx1250_TDM.h>)
  v8i z8 = {0, 0, 0, 0, 0, 0, 0, 0};
  __builtin_amdgcn_tensor_load_to_lds(g0, g1, z4, z4, z8, 0);
#else
  __builtin_amdgcn_tensor_load_to_lds(g0, g1, z4, z4, 0);
#endif
}

// ---------------------------------------------------------------------------
// Zero the one-hot output (72 MB) with coalesced float4 stores.
// ---------------------------------------------------------------------------
__global__ void zero_kernel(float* __restrict__ out, size_t n) {
  size_t i = (size_t)blockIdx.x * blockDim.x + threadIdx.x;
  size_t stride = (size_t)gridDim.x * blockDim.x;
  float4* o4 = (float4*)out;
  size_t n4 = n >> 2;
  float4 z = make_float4(0.f, 0.f, 0.f, 0.f);
  for (size_t k = i; k < n4; k += stride) o4[k] = z;
}

// ---------------------------------------------------------------------------
// Per-sample lengths via WMMA reduction: with B = all-ones,
// D[i][n] = sum_k A[i][k] + C, so sum(all 256 D elems) = 16 * sum(all A).
// Layout-agnostic and exact (mask in {0,1}, sums <= 1536, /16 exact).
// One wave per sample; EXEC all ones as WMMA requires.
// ---------------------------------------------------------------------------
__device__ __forceinline__ float wave_reduce_sum(float v) {
#pragma unroll
  for (int off = 16; off > 0; off >>= 1) v += __shfl_xor(v, off, 32);
  return v;
}

__global__ __launch_bounds__(32) void lengths_kernel(const float* __restrict__ mask,
                                                     int* __restrict__ tl,
                                                     int* __restrict__ ml) {
  const int b = blockIdx.x;
  const int lane = threadIdx.x;
  const float* mb = mask + (size_t)b * T * M;

  v2f ones;
  ones[0] = 1.0f;
  ones[1] = 1.0f;
  v8f accM = {};
  v8f accT = {};

  // m_len = sum_j mask[b][0][j] : 1536 contiguous = 24 tiles of 64
  for (int tile = 0; tile < M / 64; ++tile) {
    v2f a;
    a[0] = mb[tile * 64 + lane];
    a[1] = mb[tile * 64 + 32 + lane];
    accM = __builtin_amdgcn_wmma_f32_16x16x4_f32(false, a, false, ones, (short)0,
                                                 accM, false, false);
  }
  // t_len = sum_t mask[b][t][0] : 384 stride-M = 6 tiles of 64
  for (int tile = 0; tile < T / 64; ++tile) {
    v2f a;
    a[0] = mb[(size_t)(tile * 64 + lane) * M];
    a[1] = mb[(size_t)(tile * 64 + 32 + lane) * M];
    accT = __builtin_amdgcn_wmma_f32_16x16x4_f32(false, a, false, ones, (short)0,
                                                 accT, false, false);
  }

  float sM = 0.f, sT = 0.f;
#pragma unroll
  for (int i = 0; i < 8; ++i) {
    sM += accM[i];
    sT += accT[i];
  }
  sM = wave_reduce_sum(sM) * (1.0f / 16.0f);
  sT = wave_reduce_sum(sT) * (1.0f / 16.0f);
  if (lane == 0) {
    ml[b] = (int)(sM + 0.5f);
    tl[b] = (int)(sT + 0.5f);
  }
}

// ---------------------------------------------------------------------------
// Forward Viterbi: one wave per sample. Lane L owns rows {k*32+L}. The only
// cross-lane dependency (Q[t-1, j-1]) is a rotate-by-1 shuffle of the
// previous column -> no barriers. Columns staged into LDS by TDM, double
// buffered, pipelined with s_wait_tensorcnt. Decision bits packed 32/word.
// ---------------------------------------------------------------------------
__global__ __launch_bounds__(32) void mas_forward(const float* __restrict__ logp,
                                                  const float* __restrict__ mask,
                                                  uint32_t* __restrict__ dec,
                                                  const int* __restrict__ tlv,
                                                  const int* __restrict__ mlv) {
  const int b = blockIdx.x;
  const int lane = threadIdx.x;  // 0..31
  const int t_len = tlv[b];
  const int m_len = mlv[b];

  __shared__ float xs[2][T][TILE_C];
  __shared__ float ms[2][T][TILE_C];

  float Q[K];
  uint32_t bits[K];
#pragma unroll
  for (int k = 0; k < K; ++k) { Q[k] = NEGV; bits[k] = 0u; }

  const size_t base = (size_t)b * T * M;
  const float* lpB = logp + base;
  const float* mkB = mask + base;
  const v8i g1 = tdm_make_g1();

  // Prologue: stage chunk 0 (log_p tile + mask tile)
  tdm_issue(lpB, (uint32_t)(uintptr_t)&xs[0][0][0], g1);
  tdm_issue(mkB, (uint32_t)(uintptr_t)&ms[0][0][0], g1);

  const int nch = M / TILE_C;
  for (int c = 0; c < nch; ++c) {
    const int buf = c & 1;
    if (c + 1 < nch) {
      // Issue next chunk's DMA, then wait only for the current chunk
      // (tensor ops complete in order -> tensorcnt<=2 means first pair done).
      tdm_issue(lpB + (size_t)(c + 1) * TILE_C, (uint32_t)(uintptr_t)&xs[buf ^ 1][0][0], g1);
      tdm_issue(mkB + (size_t)(c + 1) * TILE_C, (uint32_t)(uintptr_t)&ms[buf ^ 1][0][0], g1);
      __builtin_amdgcn_s_wait_tensorcnt(2);
    } else {
      __builtin_amdgcn_s_wait_tensorcnt(0);
    }
    asm volatile("" ::: "memory");  // TDM wrote LDS behind the compiler's back

    // Pull this lane's 12 rows x 4 columns from LDS (b128 reads)
    float4 xv[K], mv[K];
#pragma unroll
    for (int k = 0; k < K; ++k) {
      xv[k] = *(const float4*)&xs[buf][k * 32 + lane][0];
      mv[k] = *(const float4*)&ms[buf][k * 32 + lane][0];
    }

#pragma unroll
    for (int cc = 0; cc < TILE_C; ++cc) {
      const int j = c * TILE_C + cc;
      // rotate previous column down one row: rot[k](L) = Q[k](L-1 mod 32)
      float rot[K];
#pragma unroll
      for (int k = 0; k < K; ++k) rot[k] = __shfl(Q[k], (lane + 31) & 31, 32);
      const uint32_t jb = 1u << (j & 31);
#pragma unroll
      for (int k = 0; k < K; ++k) {
        const int t = k * 32 + lane;
        const float sh = (lane == 0) ? ((k == 0) ? NEGV : rot[k - 1]) : rot[k];
        const float cQ = (t == j) ? NEGV : Q[k];
        const float pQ = (t == 0) ? ((j == 0) ? 0.0f : NEGV) : sh;
        const float x = ((const float*)&xv[k])[cc] * ((const float*)&mv[k])[cc];
        const float q = x + fmaxf(cQ, pQ);
        const bool valid = (t <= j) && (t >= t_len - m_len + j) && (t < t_len) && (j < m_len);
        // backtrack decision: (t==j) || Q[t,j-1] < Q[t-1,j-1]  (t==0 -> false cmp)
        const bool pred = (t == j) || ((t != 0) && (Q[k] < sh));
        if (pred) bits[k] |= jb;
        Q[k] = valid ? q : NEGV;
      }
      if ((j & 31) == 31) {  // flush packed decision words, coalesced
        const size_t db = ((size_t)b * JC + (size_t)(j >> 5)) * T;
#pragma unroll
        for (int k = 0; k < K; ++k) {
          dec[db + k * 32 + lane] = bits[k];
          bits[k] = 0u;
        }
      }
    }
  }
}

// ---------------------------------------------------------------------------
// Backtrack: wave per sample, double buffered. Lanes issue the *next* chunk's
// 12 coalesced decision-word loads (addresses independent of t) before lane 0
// walks the current chunk from LDS, hiding global latency behind the serial
// chain; lane 0 scatters the ~1536 one-hot entries.
// ---------------------------------------------------------------------------
__global__ __launch_bounds__(32) void mas_backtrack(const uint32_t* __restrict__ dec,
                                                    const int* __restrict__ tlv,
                                                    const int* __restrict__ mlv,
                                                    float* __restrict__ out) {
  const int b = blockIdx.x;
  const int lane = threadIdx.x;
  const int t_len = tlv[b];
  const int m_len = mlv[b];
  __shared__ uint32_t w[2][T];
  int t = t_len - 1;
  if (t < 0) t = 0;
  float* outB = out + (size_t)b * T * M;

  // Prologue: fetch the last chunk into registers
  uint32_t r[K];
  {
    const size_t db = ((size_t)b * JC + (JC - 1)) * T;
#pragma unroll
    for (int k = 0; k < K; ++k) r[k] = dec[db + k * 32 + lane];
  }

  for (int jc = JC - 1; jc >= 0; --jc) {
    const int cur = (JC - 1 - jc) & 1;
    // Publish current chunk's words to LDS
#pragma unroll
    for (int k = 0; k < K; ++k) w[cur][k * 32 + lane] = r[k];
    __syncthreads();
    // Issue next chunk's loads now; latency overlaps lane 0's serial walk
    if (jc > 0) {
      const size_t db = ((size_t)b * JC + (jc - 1)) * T;
#pragma unroll
      for (int k = 0; k < K; ++k) r[k] = dec[db + k * 32 + lane];
    }
    if (lane == 0) {
      for (int cc = 31; cc >= 0; --cc) {
        const int j = jc * 32 + cc;
        if (j < m_len) {
          outB[(size_t)t * M + j] = 1.0f;                      // one-hot at (t, j)
          const bool d = (t == j) || ((w[cur][t] >> cc) & 1u); // advance token?
          if (d && t > 0) --t;
        }
      }
    }
    __syncthreads();
  }
}

// ---------------------------------------------------------------------------
extern "C" void kernel_launch(void* const* d_in, const int* in_sizes, int n_in,
                              void* d_out, int out_size, void* d_ws, size_t ws_size,
                              hipStream_t stream) {
  const float* logp = (const float*)d_in[0];
  const float* mask = (const float*)d_in[1];
  float* out = (float*)d_out;

  // Workspace layout
  uint8_t* ws = (uint8_t*)d_ws;
  uint32_t* dec = (uint32_t*)ws;                       // B*JC*T words = ~2.3 MB
  int* tl = (int*)(ws + (size_t)B * JC * T * sizeof(uint32_t));
  int* ml = tl + B;

  const size_t n = (size_t)B * T * M;
  zero_kernel<<<8192, 256, 0, stream>>>(out, n);
  lengths_kernel<<<B, 32, 0, stream>>>(mask, tl, ml);
  mas_forward<<<B, 32, 0, stream>>>(logp, mask, dec, tl, ml);
  mas_backtrack<<<B, 32, 0, stream>>>(dec, tl, ml, out);
}